// MRPCEN_42949673607
// MI455X (gfx1250) — compile-verified
//
#include <hip/hip_runtime.h>
#include <hip/hip_bf16.h>
#include <stdint.h>

// MR-PCEN on MI455X (gfx1250):
//  kernel 1: per-(series,segment) EMA partial sums (wave32 butterfly reductions)
//  kernel 2: chunked scan via V_WMMA_F32_16X16X4_F32 (D = W x X + a^{i+1} carry),
//            x staged through LDS with global_load_async_to_lds_b128 (double buffer),
//            PCEN transcendentals on the VALU, coalesced b128 stores.
// Wave-uniform control goes through readfirstlane (scalar branches, EXEC all-ones
// across WMMA). The 4 t-chains are explicitly interleaved (p-outer/t-inner WMMA,
// v-outer/t-inner transcendentals) so trans/WMMA latency is hidden by ILP.
// ws usage: 4096 series * 4 t * 8 segs * 4B = 512 KB.

typedef float v2f __attribute__((ext_vector_type(2)));
typedef float v8f __attribute__((ext_vector_type(8)));

#define F_DIM   128
#define T_DIM   2000
#define EPS_F   1e-5f

__device__ __forceinline__ float wave_sum(float v) {
#pragma unroll
  for (int o = 16; o > 0; o >>= 1) v += __shfl_xor(v, o, 32);
  return v;
}

__device__ __forceinline__ void compute_sa(float* s, float* a) {
  const float tv[4] = {4.f, 16.f, 64.f, 256.f};
#pragma unroll
  for (int t = 0; t < 4; ++t) {
    float tt = tv[t] * tv[t];
    s[t] = (sqrtf(1.f + 4.f * tt) - 1.f) / (2.f * tt);
    a[t] = 1.f - s[t];
  }
}

// ---------------- kernel 1: segment partial EMA sums --------------------
// localB[series][t][seg] = sum_{k=0}^{L-1} s_t * a_t^{L-1-k} * x[series, seg_start+k]
__global__ __launch_bounds__(256) void mrpcen_seg_partials(
    const float* __restrict__ x, float* __restrict__ ws) {
  const int lane = threadIdx.x & 31;
  const int w    = __builtin_amdgcn_readfirstlane(threadIdx.x >> 5);
  const int gw   = __builtin_amdgcn_readfirstlane(blockIdx.x * 8 + w);
  const int series = gw >> 3;                 // uniform (SGPR)
  const int seg    = gw & 7;
  const int segmin = (seg < 5) ? seg : 5;
  const int c_start = seg * 15 + segmin;      // chunk index
  const int nch     = (seg < 5) ? 16 : 15;
  const int L       = nch * 16;

  float s[4], a[4];
  compute_sa(s, a);
  float aw[4], a32[4], acc[4];
#pragma unroll
  for (int t = 0; t < 4; ++t) {
    aw[t]  = s[t] * __powf(a[t], (float)(31 - lane));
    a32[t] = __powf(a[t], 32.f);
    acc[t] = 0.f;
  }
  const float* xp = x + series * T_DIM + c_start * 16;
  int base = 0;
  for (; base + 32 <= L; base += 32) {
    float xv = xp[base + lane];
#pragma unroll
    for (int t = 0; t < 4; ++t) {
      float red = wave_sum(aw[t] * xv);
      acc[t] = acc[t] * a32[t] + red;
    }
  }
  if (base < L) {  // 16-element tail (segments with 15 chunks); uniform branch
    float xv = (lane < 16) ? xp[base + lane] : 0.f;
#pragma unroll
    for (int t = 0; t < 4; ++t) {
      float wt  = (lane < 16) ? s[t] * __powf(a[t], (float)(15 - lane)) : 0.f;
      float red = wave_sum(wt * xv);
      acc[t] = acc[t] * __powf(a[t], 16.f) + red;
    }
  }
  if (lane == 0) {
#pragma unroll
    for (int t = 0; t < 4; ++t) ws[(series * 4 + t) * 8 + seg] = acc[t];
  }
}

// ---------------- kernel 2: WMMA chunk-scan + PCEN ----------------------
__global__ __launch_bounds__(128) void mrpcen_main(
    const float* __restrict__ x, const float* __restrict__ log_alpha,
    const float* __restrict__ log_delta, const float* __restrict__ log_r,
    float* __restrict__ out, const float* __restrict__ ws) {
  // per-wave double buffer: 16 rows x 20 floats (80B row: conflict-free, 16B aligned)
  __shared__ __align__(16) float smem[4][2][16][20];

  const int lane = threadIdx.x & 31;
  const int w    = __builtin_amdgcn_readfirstlane(threadIdx.x >> 5);
  const int gw   = __builtin_amdgcn_readfirstlane(blockIdx.x * 4 + w);
  const int tile = gw >> 3;                  // all uniform (SGPR)
  const int seg  = gw & 7;
  const int f      = tile >> 1;
  const int b_base = (tile & 1) * 16;
  const int n    = lane & 15;                // matrix column = batch row
  const int half = lane >> 4;

  const int segmin  = (seg < 5) ? seg : 5;
  const int c_start = seg * 15 + segmin;
  const int nch     = (seg < 5) ? 16 : 15;

  float s[4], a[4];
  compute_sa(s, a);
  const float alpha  = __expf(log_alpha[f]);   // f uniform -> scalar loads
  const float delta  = __expf(log_delta[f]);
  const float rr     = __expf(log_r[f]);
  const float deltar = __powf(delta, rr);

  const int series = (b_base + n) * F_DIM + f;

  // exact carry at segment entry (carry == M[seg_start-1]; M[-1] := x0 so M[0]=x0)
  float carry[4];
  {
    float x0 = x[series * T_DIM];
#pragma unroll
    for (int t = 0; t < 4; ++t) carry[t] = x0;
    for (int j = 0; j < seg; ++j) {          // uniform trip count
      float Lj = (j < 5) ? 256.f : 240.f;
#pragma unroll
      for (int t = 0; t < 4; ++t)
        carry[t] = carry[t] * __powf(a[t], Lj) + ws[(series * 4 + t) * 8 + j];
    }
  }

  // C base scale: C[i,n] = a^{i+1}*carry[n], row i = v + 8*half.
  // Only keep a^{(8*half)+1}; the v-direction is rebuilt iteratively each chunk.
  float apb[4];
#pragma unroll
  for (int t = 0; t < 4; ++t) apb[t] = half ? __powf(a[t], 9.f) : a[t];

  // A pieces: W[i,kg] = (i>=kg) ? s*a^{i-kg} : 0 ; A 16x4 layout: K = p*4 + 2*half + e
  v2f wA[4][4];
#pragma unroll
  for (int t = 0; t < 4; ++t) {
#pragma unroll
    for (int p = 0; p < 4; ++p) {
      int k0 = p * 4 + 2 * half;
      float e0 = (n >= k0)     ? s[t] * __powf(a[t], (float)(n - k0))     : 0.f;
      float e1 = (n >= k0 + 1) ? s[t] * __powf(a[t], (float)(n - k0 - 1)) : 0.f;
      v2f wv = {e0, e1};
      wA[t][p] = wv;
    }
  }

  const uint32_t sbase = (uint32_t)(uintptr_t)(&smem[w][0][0][0]);
  const uint64_t xbase = (uint64_t)(uintptr_t)x;
  const uint32_t grow  = (uint32_t)(series * T_DIM) * 4u;

  // async stage of one 16x16 x-tile (2 x b128 per lane = 1024B) into buffer `buf`
  auto stage = [&](int ci, int buf) {
    const uint32_t tau = (uint32_t)(c_start + ci) * 16u;
    uint32_t g0 = grow + tau * 4u + (uint32_t)half * 16u;
    uint32_t l0 = sbase + (uint32_t)buf * 1280u + (uint32_t)n * 80u +
                  (uint32_t)half * 16u;
    asm volatile(
        "s_wait_dscnt 0x0\n\t"
        "global_load_async_to_lds_b128 %0, %1, %4\n\t"
        "global_load_async_to_lds_b128 %2, %3, %4"
        :: "v"(l0), "v"(g0), "v"(l0 + 32u), "v"(g0 + 32u), "s"(xbase)
        : "memory");
  };

  stage(0, 0);
  if (nch > 1) stage(1, 1);

  for (int ci = 0; ci < nch; ++ci) {         // uniform loop
    const int buf = ci & 1;
    if (ci < nch - 1) asm volatile("s_wait_asynccnt 0x2" ::: "memory");
    else              asm volatile("s_wait_asynccnt 0x0" ::: "memory");

    const float* row = &smem[w][buf][n][0];
    // B 4x16 pieces: B[klocal, n] = x[n, tau + p*4 + klocal]; K = 2*half + e
    v2f B[4];
#pragma unroll
    for (int p = 0; p < 4; ++p) {
      const float2 bb = *(const float2*)(row + p * 4 + 2 * half);
      v2f bv = {bb.x, bb.y};
      B[p] = bv;
    }
    // x needed again for PCEN, already in D layout: xr[v] = x[n, tau + v + 8*half]
    const float4 xa = *(const float4*)(row + half * 8);
    const float4 xb = *(const float4*)(row + half * 8 + 4);
    float xr[8] = {xa.x, xa.y, xa.z, xa.w, xb.x, xb.y, xb.z, xb.w};

    if (ci + 2 < nch) stage(ci + 2, buf);    // uniform prefetch branch

    // ---- C operands: d4[t][v] = a^{v+8*half+1} * carry[t] (4 indep chains) ----
    v8f d4[4];
#pragma unroll
    for (int t = 0; t < 4; ++t) {
      d4[t][0] = apb[t] * carry[t];
#pragma unroll
      for (int v = 1; v < 8; ++v) d4[t][v] = d4[t][v - 1] * a[t];
    }
    // ---- WMMA: p-outer / t-inner => 4 independent v_wmma per step ----
#pragma unroll
    for (int p = 0; p < 4; ++p) {
#pragma unroll
      for (int t = 0; t < 4; ++t)
        d4[t] = __builtin_amdgcn_wmma_f32_16x16x4_f32(
            false, wA[t][p], false, B[p], (short)0, d4[t], false, false);
    }
    // carry = M row 15 = D VGPR7, lanes 16..31 (N = lane-16); before overwrite
#pragma unroll
    for (int t = 0; t < 4; ++t) carry[t] = __shfl(d4[t][7], 16 + n, 32);

    // ---- PCEN: v-outer / t-inner => 4 independent log/exp chains per step ----
#pragma unroll
    for (int v = 0; v < 8; ++v) {
#pragma unroll
      for (int t = 0; t < 4; ++t) {
        // smooth = (eps+M)^(-alpha);  pcen = (x*smooth+delta)^r - delta^r
        float sm  = __builtin_amdgcn_exp2f(
            -alpha * __builtin_amdgcn_logf(EPS_F + d4[t][v]));
        float bse = xr[v] * sm + delta;
        d4[t][v] = __builtin_amdgcn_exp2f(rr * __builtin_amdgcn_logf(bse)) - deltar;
      }
    }

    const uint32_t tau = (uint32_t)(c_start + ci) * 16u;
#pragma unroll
    for (int t = 0; t < 4; ++t) {
      uint32_t ob = ((uint32_t)(((b_base + n) * 4 + t) * F_DIM + f)) * (uint32_t)T_DIM +
                    tau + (uint32_t)half * 8u;
      *(float4*)(out + ob)     = make_float4(d4[t][0], d4[t][1], d4[t][2], d4[t][3]);
      *(float4*)(out + ob + 4) = make_float4(d4[t][4], d4[t][5], d4[t][6], d4[t][7]);
    }
  }
}

extern "C" void kernel_launch(void* const* d_in, const int* in_sizes, int n_in,
                              void* d_out, int out_size, void* d_ws, size_t ws_size,
                              hipStream_t stream) {
  const float* x  = (const float*)d_in[0];
  const float* la = (const float*)d_in[1];
  const float* ld = (const float*)d_in[2];
  const float* lr = (const float*)d_in[3];
  float* out = (float*)d_out;
  float* ws  = (float*)d_ws;   // needs 4096*4*8*4 = 512 KB

  // kernel 1: 4096 series * 8 segs = 32768 waves, 8 waves/block
  mrpcen_seg_partials<<<4096, 256, 0, stream>>>(x, ws);
  // kernel 2: 256 tiles * 8 segs = 2048 waves, 4 waves/block
  mrpcen_main<<<512, 128, 0, stream>>>(x, la, ld, lr, out, ws);
}